// FuncToNode_60730837565801
// MI455X (gfx1250) — compile-verified
//
#include <hip/hip_runtime.h>
#include <math.h>

typedef __attribute__((ext_vector_type(2))) float v2f;
typedef __attribute__((ext_vector_type(8))) float v8f;

#define N_F   1024
#define N_N   2048
#define DIM   64
#define NBATCH 16
#define KTOT  768            // 256 features * 3 scales
#define EPSV  1e-6f
#define LN_EPS 1e-5f

// ---- workspace layout (float offsets) ----
#define WS_FEAT 0                        // 2048 * 256
#define WS_SLOG (N_N * 256)              // 2048   (log degree)
#define WS_SCL  (WS_SLOG + N_N)          // 2048*2 (scale, 1/scale)
#define WS_BSUM (WS_SCL + 2 * N_N)       // 16
#define WS_BCNT (WS_BSUM + NBATCH)       // 16

// ---- optional CDNA5 async global->LDS path (guarded) ----
#if defined(__has_builtin)
#if __has_builtin(__builtin_amdgcn_global_load_async_to_lds_b128)
#define HAVE_ASYNC_COPY 1
#endif
#if __has_builtin(__builtin_amdgcn_s_wait_asynccnt)
#define HAVE_WAIT_ASYNC 1
#endif
#endif
#ifndef HAVE_ASYNC_COPY
#define HAVE_ASYNC_COPY 0
#endif
#ifndef HAVE_WAIT_ASYNC
#define HAVE_WAIT_ASYNC 0
#endif

#if HAVE_ASYNC_COPY
// builtin expects pointers to 16-byte int vectors (per hipcc diagnostic):
//   arg0: global (AS1) v4i*   arg1: LDS (AS3) v4i*   arg2/3: imm offset, cpol
typedef int v4i_vec __attribute__((vector_size(16)));
typedef __attribute__((address_space(1))) v4i_vec* g_v4i_p;
typedef __attribute__((address_space(3))) v4i_vec* l_v4i_p;
#endif

// ------------------------------------------------------------------
// K0: zero the batch accumulators (32 floats)
// ------------------------------------------------------------------
__global__ void k0_zero(float* __restrict__ ws) {
    int t = threadIdx.x;
    if (t < 2 * NBATCH) ws[WS_BSUM + t] = 0.0f;
}

// ------------------------------------------------------------------
// K1: per-node aggregation (sum / sumsq / min / max / degree)
// One block per node, lane d owns dim d. min/max force a full scan of
// A's column, so sum/sumsq/degree ride the same pass. The column walk
// is stride-8KB, so prefetch the A cacheline 32 rows ahead
// (global_prefetch_b8) to keep the latency-bound stream fed.
// ------------------------------------------------------------------
__global__ __launch_bounds__(64)
void k1_agg(const float* __restrict__ A, const float* __restrict__ x,
            const int* __restrict__ bn, float* __restrict__ ws) {
    const int n = blockIdx.x;
    const int d = threadIdx.x;

    float sum = 0.0f, sq = 0.0f, dg = 0.0f;
    float mn = INFINITY, mx = -INFINITY;

#pragma unroll 4
    for (int f = 0; f < N_F; ++f) {
        if ((f & 3) == 0) {
            int fp = f + 32 < N_F ? f + 32 : N_F - 1;
            __builtin_prefetch(A + (size_t)fp * N_N + n, 0, 1);
        }
        float a  = A[(size_t)f * N_N + n];   // uniform across the block
        float xv = x[(size_t)f * DIM + d];   // coalesced
        sum = fmaf(a, xv, sum);
        sq  = fmaf(a, xv * xv, sq);
        dg += a;
        bool on = (a != 0.0f);
        mn = fminf(mn, on ? xv :  INFINITY);
        mx = fmaxf(mx, on ? xv : -INFINITY);
    }

    float deg  = dg + 1.0f;
    float cd   = fmaxf(deg, EPSV);
    float mean = sum / cd;
    float sqm  = sq / cd;
    float sd   = sqrtf(fmaxf(sqm - mean * mean, EPSV));

    float* feat = ws + WS_FEAT + (size_t)n * 256;
    feat[d]       = mean;
    feat[64 + d]  = mn;
    feat[128 + d] = mx;
    feat[192 + d] = sd;

    if (d == 0) {
        float lg = logf(deg);
        ws[WS_SLOG + n] = lg;
        int bb = bn[n];
        atomicAdd(&ws[WS_BSUM + bb], lg);
        atomicAdd(&ws[WS_BCNT + bb], 1.0f);
    }
}

// ------------------------------------------------------------------
// K2: per-node degree scales from batch segment means
// ------------------------------------------------------------------
__global__ void k2_scale(const int* __restrict__ bn, float* __restrict__ ws) {
    int n = blockIdx.x * blockDim.x + threadIdx.x;
    if (n >= N_N) return;
    int bb   = bn[n];
    float ms = ws[WS_BSUM + bb] / fmaxf(ws[WS_BCNT + bb], EPSV);
    float sc = ws[WS_SLOG + n] / fmaxf(ms, EPSV);
    ws[WS_SCL + 2 * n]     = sc;
    ws[WS_SCL + 2 * n + 1] = 1.0f / fmaxf(sc, EPSV);
}

// ------------------------------------------------------------------
// K3: fused WMMA f32 GEMM (2048x768 @ 768x64) + bias + LayerNorm + ReLU
//
// Block = 128 threads = 4 waves; wave w owns the 16x16 D tile for dims
// [16w,16w+16) of 16 nodes (all 64 dims of a node stay in one block so
// the LN row-reduction is local).
//
// Phase 1: async-copy the 16x256 feature tile (16KB) global->LDS.
// Phase 2: expand to the 16x768 'update' tile in LDS, stored k-major
//          (upd[k*16+m]) so A-frag reads hit 16 distinct banks.
// Phase 3: WMMA loop: ds_load A frag + global float2 B frag (W rows,
//          L2-resident) + v_wmma_f32_16x16x4_f32 — no per-iter ALU.
//
// f32 WMMA 16x16x4 VGPR striping (ISA 7.12.2):
//   A (16x4): lane m=L%16; VGPR0 holds K=(L<16?0:2), VGPR1 K=(L<16?1:3)
//   B (4x16): column N=L%16; VGPR0 holds K=(L<16?0:2), VGPR1 K=(L<16?1:3)
//   D (16x16): VGPR r -> row r + 8*(L/16), col L%16
//
// LDS: upd2 48KB + scr 16KB = 64KB. scr holds the feature staging tile
// during phases 1-2, then is reused for the LN scratch (h tile +
// partials) after the post-phase-2 barrier makes features dead.
// ------------------------------------------------------------------
__global__ __launch_bounds__(128)
void k3_gemm_ln(const float* __restrict__ W, const float* __restrict__ bias,
                const float* __restrict__ gamma, const float* __restrict__ beta,
                const float* __restrict__ ws, float* __restrict__ out) {
    __shared__ float upd2[KTOT * 16];   // k-major update tile: [k][m]
    __shared__ float scr[4096];         // feat staging, then LN scratch

    const int n0   = blockIdx.x * 16;
    const int t    = threadIdx.x;
    const int lane = t & 31;
    const int wv   = t >> 5;            // 0..3 -> dim tile
    const int d0   = wv * 16;
    const int m    = lane & 15;         // node row within tile
    const int koff = (lane >> 4) << 1;  // 0 or 2 (K sub-step)
    const int dcol = d0 + m;            // this lane's B/D column

    // ---- phase 1: stage the 16x256 feature tile (contiguous 16KB) ----
    const float* gfeat = ws + WS_FEAT + (size_t)n0 * 256;   // 4096 floats
#pragma unroll
    for (int i = t; i < 1024; i += 128) {
#if HAVE_ASYNC_COPY
        __builtin_amdgcn_global_load_async_to_lds_b128(
            (g_v4i_p)(const_cast<float*>(gfeat + i * 4)),
            (l_v4i_p)(&scr[i * 4]), 0, 0);
#else
        float4 v = *(const float4*)(gfeat + i * 4);
        *(float4*)(&scr[i * 4]) = v;
#endif
    }
#if HAVE_ASYNC_COPY
#if HAVE_WAIT_ASYNC
    __builtin_amdgcn_s_wait_asynccnt(0);
#else
    asm volatile("s_wait_asynccnt 0x0" ::: "memory");
#endif
#endif
    __syncthreads();

    // ---- phase 2: expand features -> update tile (k-major) ----
    {
        const int em  = t >> 3;                 // 0..15 node row
        const int eq0 = t & 7;                  // feature phase
        const float es2 = ws[WS_SCL + 2 * (n0 + em)];
        const float es3 = ws[WS_SCL + 2 * (n0 + em) + 1];
#pragma unroll
        for (int it = 0; it < 32; ++it) {
            int q = eq0 + (it << 3);            // 0..255
            float f = scr[em * 256 + q];
            int kb = q * 3;
            upd2[(kb    ) * 16 + em] = f;
            upd2[(kb + 1) * 16 + em] = f * es2;
            upd2[(kb + 2) * 16 + em] = f * es3;
        }
    }
    __syncthreads();   // features in scr are dead past this point

    // ---- phase 3: WMMA main loop ----
    const float*  updcol = upd2 + m;
    const float2* wrow   = (const float2*)(W + (size_t)dcol * KTOT);

    v8f acc = {0.f, 0.f, 0.f, 0.f, 0.f, 0.f, 0.f, 0.f};

#pragma unroll 4
    for (int k0 = 0; k0 < KTOT; k0 += 4) {
        const int kA = k0 + koff;
        v2f afrag;
        afrag.x = updcol[(kA    ) * 16];
        afrag.y = updcol[(kA + 1) * 16];
        float2 wb = wrow[(unsigned)kA >> 1];    // 8B-aligned
        v2f bfrag;
        bfrag.x = wb.x;
        bfrag.y = wb.y;
        acc = __builtin_amdgcn_wmma_f32_16x16x4_f32(
            /*neg_a=*/false, afrag, /*neg_b=*/false, bfrag,
            /*c_mod=*/(short)0, acc, /*reuse_a=*/false, /*reuse_b=*/false);
    }

    // ---- epilogue: bias + LayerNorm + ReLU (scratch overlaid in scr) ----
    float* hs   = scr;            // 16 x 66 = 1056 floats
    float* ps1  = scr + 1056;     // 16 x 8
    float* ps2  = scr + 1184;     // 16 x 8
    float* mu_s = scr + 1312;     // 16
    float* rs_s = scr + 1328;     // 16

    const float bv    = bias[dcol];
    const int   mbase = (lane >> 4) << 3;   // 0 or 8
#pragma unroll
    for (int r = 0; r < 8; ++r)
        hs[(mbase + r) * 66 + dcol] = acc[r] + bv;
    __syncthreads();

    const int nl = t & 15;
    const int g  = t >> 4;        // 0..7
    const int cb = g * 8;
    float a1 = 0.0f, a2 = 0.0f;
#pragma unroll
    for (int i = 0; i < 8; ++i) {
        float v = hs[nl * 66 + cb + i];
        a1 += v;
        a2 += v * v;
    }
    ps1[nl * 8 + g] = a1;
    ps2[nl * 8 + g] = a2;
    __syncthreads();

    if (t < 16) {
        float S1 = 0.0f, S2 = 0.0f;
#pragma unroll
        for (int gg = 0; gg < 8; ++gg) {
            S1 += ps1[t * 8 + gg];
            S2 += ps2[t * 8 + gg];
        }
        float mu  = S1 * (1.0f / 64.0f);
        float var = S2 * (1.0f / 64.0f) - mu * mu;
        mu_s[t] = mu;
        rs_s[t] = rsqrtf(var + LN_EPS);
    }
    __syncthreads();

    const float mu   = mu_s[nl];
    const float rstd = rs_s[nl];
#pragma unroll
    for (int i = 0; i < 8; ++i) {
        int d   = cb + i;
        float v = hs[nl * 66 + d];
        float o = (v - mu) * rstd * gamma[d] + beta[d];
        out[(size_t)(n0 + nl) * DIM + d] = fmaxf(o, 0.0f);
    }
}

// ------------------------------------------------------------------
extern "C" void kernel_launch(void* const* d_in, const int* in_sizes, int n_in,
                              void* d_out, int out_size, void* d_ws, size_t ws_size,
                              hipStream_t stream) {
    const float* A     = (const float*)d_in[0];   // (1024, 2048)
    const float* x     = (const float*)d_in[1];   // (1024, 64)
    const float* W     = (const float*)d_in[2];   // (64, 768) row-major
    const float* b     = (const float*)d_in[3];   // (64,)
    const float* gamma = (const float*)d_in[4];   // (64,)
    const float* beta  = (const float*)d_in[5];   // (64,)
    const int*   bn    = (const int*)d_in[6];     // (2048,)
    float* ws  = (float*)d_ws;
    float* out = (float*)d_out;

    (void)in_sizes; (void)n_in; (void)out_size; (void)ws_size;

    hipLaunchKernelGGL(k0_zero,   dim3(1),         dim3(32),  0, stream, ws);
    hipLaunchKernelGGL(k1_agg,    dim3(N_N),       dim3(64),  0, stream, A, x, bn, ws);
    hipLaunchKernelGGL(k2_scale,  dim3(N_N / 256), dim3(256), 0, stream, bn, ws);
    hipLaunchKernelGGL(k3_gemm_ln, dim3(N_N / 16), dim3(128), 0, stream,
                       W, b, gamma, beta, ws, out);
}